// TorchSTFT_15865609191759
// MI455X (gfx1250) — compile-verified
//
#include <hip/hip_runtime.h>
#include <hip/hip_bf16.h>
#include <math.h>

// ---------------- problem constants ----------------
#define HOP      200
#define WINL     800          // window / DFT length N
#define NBINS    401          // N/2+1
#define NPAD     416          // NBINS padded to 26*16 for WMMA N-tiles
#define KINV     404          // NBINS padded to mult of 4 (101 K-steps)
#define NTIME    160000
#define NBATCH   32
#define NFRAMES  801          // (160800-800)/200+1
#define RECLEN   160800       // (F-1)*HOP + WIN
#define PADL     400          // reflect pad = N/2

#define SFWD     200          // forward K-steps  (800/4)
#define SINV     101          // inverse K-steps  (404/4)
#define WFS      801          // LDS stride (odd -> no bank conflicts)
#define SRS      417          // LDS stride for sre/sim

// workspace layout (floats)
#define NF_PACK  (SFWD*NPAD*4)    // 332800 per fwd matrix
#define NI_PACK  (SINV*WINL*4)    // 323200 per inv matrix
#define OFF_BFC  0
#define OFF_BFS  (OFF_BFC + NF_PACK)
#define OFF_BIC  (OFF_BFS + NF_PACK)
#define OFF_BIS  (OFF_BIC + NI_PACK)
#define OFF_REC  (OFF_BIS + NI_PACK)
#define WS_FLOATS (OFF_REC + (size_t)NBATCH*RECLEN)   // ~6.46M floats ~ 25.8 MB

typedef __attribute__((ext_vector_type(2))) float v2f;
typedef __attribute__((ext_vector_type(8))) float v8f;

#define TWO_PI_OVER_N 0.007853981633974483f   // 2*pi/800

// ------------------------------------------------------------------
// Build WMMA-packed DFT matrices.
//   Forward:  B[n][k] packed as pack[s][k][j], row = 4s+j = n (sample),
//             col = k (bin).  cos part = cos(2pi*k*n/N); sin part = -sin(...).
//   Inverse:  B[k][n] packed as pack[s][n][j], row = 4s+j = k (bin),
//             col = n (sample). icos = w_k*cos/N ; isin = -w_k*sin/N,
//             w_k = 1 for k==0,400 else 2 (Hermitian fold).  k>400 -> 0.
// Integer reduction (k*n)%N keeps angles exact before cosf/sinf.
// ------------------------------------------------------------------
__global__ void init_mats(float* __restrict__ ws) {
    float* Bfc = ws + OFF_BFC;
    float* Bfs = ws + OFF_BFS;
    float* Bic = ws + OFF_BIC;
    float* Bis = ws + OFF_BIS;
    const int total = 2*NF_PACK + 2*NI_PACK;
    for (int idx = blockIdx.x*blockDim.x + threadIdx.x; idx < total;
         idx += gridDim.x*blockDim.x) {
        if (idx < 2*NF_PACK) {
            int i = idx;  bool isSin = false;
            if (i >= NF_PACK) { i -= NF_PACK; isSin = true; }
            int j = i & 3;
            int c = (i >> 2) % NPAD;       // bin column
            int s = i / (NPAD*4);
            int n = 4*s + j;               // sample row
            float v = 0.0f;
            if (c < NBINS) {
                int r = (c * n) % WINL;
                float ang = (float)r * TWO_PI_OVER_N;
                v = isSin ? -sinf(ang) : cosf(ang);
            }
            (isSin ? Bfs : Bfc)[(size_t)(s*NPAD + c)*4 + j] = v;
        } else {
            int i = idx - 2*NF_PACK;  bool isSin = false;
            if (i >= NI_PACK) { i -= NI_PACK; isSin = true; }
            int j = i & 3;
            int c = (i >> 2) % WINL;       // sample column
            int s = i / (WINL*4);
            int k = 4*s + j;               // bin row
            float v = 0.0f;
            if (k < NBINS) {
                float wk = (k == 0 || k == NBINS-1) ? 1.0f : 2.0f;
                int r = (k * c) % WINL;
                float ang = (float)r * TWO_PI_OVER_N;
                v = (isSin ? -sinf(ang) : cosf(ang)) * wk * (1.0f/(float)WINL);
            }
            (isSin ? Bis : Bic)[(size_t)(s*WINL + c)*4 + j] = v;
        }
    }
}

__global__ void zero_rec(float* __restrict__ rec, int n) {
    int i = blockIdx.x*blockDim.x + threadIdx.x;
    if (i < n) rec[i] = 0.0f;
}

// ------------------------------------------------------------------
// Fused STFT -> iSTFT -> windowed overlap-add for a tile of 16 frames.
// ------------------------------------------------------------------
__global__ __launch_bounds__(256) void stft_fused(
    const float* __restrict__ x,
    const float* __restrict__ Bfc, const float* __restrict__ Bfs,
    const float* __restrict__ Bic, const float* __restrict__ Bis,
    float* __restrict__ rec)
{
    __shared__ float wf [16*WFS];     // windowed frames      (M x K=800)
    __shared__ float sre[16*SRS];     // real spectrum        (M x K<=416)
    __shared__ float sim[16*SRS];     // imag spectrum
    __shared__ float winl[WINL];      // Hann window
    __shared__ int   bofs[16];        // batch offset into rec
    __shared__ int   fpos[16];        // frame*HOP

    const int tid  = threadIdx.x;
    const int g0   = blockIdx.x * 16;          // first global frame of tile
    const int wave = tid >> 5;
    const int lane = tid & 31;
    const int ln   = lane & 15;                // N-index within 16-wide tile / M row
    const int p    = lane >> 4;                // lane-half selector

    if (tid < 16) {
        int g = g0 + tid;
        int b = g / NFRAMES;
        int f = g - b*NFRAMES;
        bofs[tid] = b * RECLEN;
        fpos[tid] = f * HOP;
    }
    for (int n = tid; n < WINL; n += 256)
        winl[n] = 0.5f - 0.5f*cosf(TWO_PI_OVER_N * (float)n);
    __syncthreads();

    // ---- stage A: gather (reflect pad) + analysis window into LDS ----
    for (int i = tid; i < 16*WINL; i += 256) {
        int m = i / WINL;
        int n = i - m*WINL;
        int g = g0 + m;
        int b = g / NFRAMES;
        int f = g - b*NFRAMES;
        int t = f*HOP + n - PADL;
        if (t < 0)       t = -t;
        if (t >= NTIME)  t = 2*(NTIME-1) - t;
        wf[m*WFS + n] = x[(size_t)b*NTIME + t] * winl[n];
    }
    __syncthreads();

    // ---- stage B: forward DFT.  re/im[16 x 416] = wf[16 x 800] * B ----
    for (int tile = wave; tile < NPAD/16; tile += 8) {
        int col = tile*16 + ln;                       // bin column
        v8f cc = {0.f,0.f,0.f,0.f,0.f,0.f,0.f,0.f};
        v8f cs = {0.f,0.f,0.f,0.f,0.f,0.f,0.f,0.f};
        const v2f*  pc = (const v2f*)(Bfc + (size_t)col*4 + 2*p);
        const v2f*  ps = (const v2f*)(Bfs + (size_t)col*4 + 2*p);
        const float* pa = wf + ln*WFS + 2*p;
        #pragma unroll 4
        for (int s = 0; s < SFWD; ++s) {
            v2f a; a.x = pa[4*s]; a.y = pa[4*s + 1];
            v2f b1 = pc[s*(NPAD*4/2)];
            v2f b2 = ps[s*(NPAD*4/2)];
            cc = __builtin_amdgcn_wmma_f32_16x16x4_f32(false, a, false, b1,
                                                       (short)0, cc, false, false);
            cs = __builtin_amdgcn_wmma_f32_16x16x4_f32(false, a, false, b2,
                                                       (short)0, cs, false, false);
        }
        #pragma unroll
        for (int r = 0; r < 8; ++r) {
            int M = r + 8*p;
            sre[M*SRS + col] = cc[r];
            sim[M*SRS + col] = cs[r];
        }
    }
    __syncthreads();

    // ---- stage C: inverse DFT + synthesis window + overlap-add ----
    const float inv_norm = sqrtf(1.0f - (float)HOP/(float)WINL + 1e-8f);
    for (int tile = wave; tile < WINL/16; tile += 8) {
        int col = tile*16 + ln;                       // sample column n
        v8f acc = {0.f,0.f,0.f,0.f,0.f,0.f,0.f,0.f};
        const v2f*  pic = (const v2f*)(Bic + (size_t)col*4 + 2*p);
        const v2f*  pis = (const v2f*)(Bis + (size_t)col*4 + 2*p);
        const float* pr = sre + ln*SRS + 2*p;
        const float* pi = sim + ln*SRS + 2*p;
        #pragma unroll 4
        for (int s = 0; s < SINV; ++s) {
            v2f a1; a1.x = pr[4*s]; a1.y = pr[4*s + 1];
            v2f b1 = pic[s*(WINL*4/2)];
            acc = __builtin_amdgcn_wmma_f32_16x16x4_f32(false, a1, false, b1,
                                                        (short)0, acc, false, false);
            v2f a2; a2.x = pi[4*s]; a2.y = pi[4*s + 1];
            v2f b2 = pis[s*(WINL*4/2)];
            acc = __builtin_amdgcn_wmma_f32_16x16x4_f32(false, a2, false, b2,
                                                        (short)0, acc, false, false);
        }
        float wsyn = winl[col] * inv_norm;
        #pragma unroll
        for (int r = 0; r < 8; ++r) {
            int M = r + 8*p;
            atomicAdd(rec + bofs[M] + fpos[M] + col, acc[r] * wsyn);
        }
    }
}

__global__ void crop_out(const float* __restrict__ rec, float* __restrict__ out) {
    int i = blockIdx.x*blockDim.x + threadIdx.x;
    int total = NBATCH * NTIME;
    if (i < total) {
        int b = i / NTIME;
        int t = i - b*NTIME;
        out[i] = rec[(size_t)b*RECLEN + t + PADL];
    }
}

extern "C" void kernel_launch(void* const* d_in, const int* in_sizes, int n_in,
                              void* d_out, int out_size, void* d_ws, size_t ws_size,
                              hipStream_t stream) {
    const float* x = (const float*)d_in[0];
    float* ws  = (float*)d_ws;          // needs ~26 MB of scratch
    float* out = (float*)d_out;

    float* Bfc = ws + OFF_BFC;
    float* Bfs = ws + OFF_BFS;
    float* Bic = ws + OFF_BIC;
    float* Bis = ws + OFF_BIS;
    float* rec = ws + OFF_REC;

    {   // 1) DFT matrices (WMMA-packed)
        int total = 2*NF_PACK + 2*NI_PACK;
        init_mats<<<(total + 255)/256, 256, 0, stream>>>(ws);
    }
    {   // 2) clear overlap-add accumulator
        int n = NBATCH*RECLEN;
        zero_rec<<<(n + 255)/256, 256, 0, stream>>>(rec, n);
    }
    {   // 3) fused STFT -> iSTFT -> OLA (16 frames per block, 25632 frames)
        int nTiles = (NBATCH*NFRAMES)/16;       // 1602, exact
        stft_fused<<<nTiles, 256, 0, stream>>>(x, Bfc, Bfs, Bic, Bis, rec);
    }
    {   // 4) crop padding into output
        int n = NBATCH*NTIME;
        crop_out<<<(n + 255)/256, 256, 0, stream>>>(rec, out);
    }
}